// DTDFCell_50714973831168
// MI455X (gfx1250) — compile-verified
//
#include <hip/hip_runtime.h>

typedef __attribute__((ext_vector_type(2))) float v2f;
typedef __attribute__((ext_vector_type(4))) float v4f;
typedef __attribute__((ext_vector_type(8))) float v8f;

#define WAVES_PER_BLOCK 8
#define TILES_PER_WAVE  4
#define TILE            16
#define SAMPLES_PER_WAVE  (TILES_PER_WAVE * TILE)            // 64
#define SAMPLES_PER_BLOCK (WAVES_PER_BLOCK * SAMPLES_PER_WAVE) // 512

// Per-wave LDS (floats): staged v tiles (4*16*8=512) + D staging (16 rows * 20) = 832
#define LDSV_F 512
#define LDSO_F 320
#define WAVE_F (LDSV_F + LDSO_F)

// T[n][k]: coefficient of input-feature k in output-feature n.
// features: 0 = x, 1..8 = v0..v7.  outputs: 0 = out, 1..8 = v_new[0..7].
__device__ __forceinline__ float tcoef(int n, int k,
                                       const float* __restrict__ b,
                                       const float* __restrict__ a) {
    if (n > 8 || k > 8) return 0.0f;
    if (n == 0) return (k == 0) ? b[0] : ((k == 1) ? 1.0f : 0.0f);
    const int m = n - 1;
    float r = 0.0f;
    if (k == 0)      r = b[m + 1] - a[m] * b[0];
    else if (k == 1) r = -a[m];
    if (k == m + 2)  r += 1.0f;   // shift term v_new[m] += v[m+1] (m<7)
    return r;
}

__global__ __launch_bounds__(256) void dtdf_wmma_kernel(
    const float* __restrict__ x_in, const float* __restrict__ v_in,
    const float* __restrict__ bc,   const float* __restrict__ ac,
    float* __restrict__ out, float* __restrict__ vnew)
{
    __shared__ float smem[WAVES_PER_BLOCK * WAVE_F];

    const int tid  = threadIdx.x;
    const int w    = tid >> 5;    // wave32
    const int l    = tid & 31;
    const int half = l >> 4;
    const int ln   = l & 15;

    float* ldsv = &smem[w * WAVE_F];
    float* ldso = ldsv + LDSV_F;

    const long long wbase = (long long)blockIdx.x * SAMPLES_PER_BLOCK
                          + (long long)w * SAMPLES_PER_WAVE;

    // ---- CDNA5 async global -> LDS staging of v (ASYNCcnt path) ----
    {
        const unsigned lds0 = (unsigned)(size_t)(const void*)ldsv;
#pragma unroll
        for (int t = 0; t < TILES_PER_WAVE; ++t) {
            unsigned loff = lds0 + (unsigned)((t * 128 + l * 4) * 4);
            unsigned long long ga =
                (unsigned long long)(const void*)(v_in + (wbase + t * TILE) * 8 + l * 4);
            asm volatile("global_load_async_to_lds_b128 %0, %1, off"
                         :: "v"(loff), "v"(ga) : "memory");
        }
        asm volatile("s_wait_asynccnt 0x0" ::: "memory");
    }

    // ---- B operand = T^T in WMMA 4x16 layout: lane = N (out feature), K = j + 2*half ----
    v2f B0, B1, B2;
    B0.x = tcoef(ln, 0 + 2 * half, bc, ac);
    B0.y = tcoef(ln, 1 + 2 * half, bc, ac);
    B1.x = tcoef(ln, 4 + 2 * half, bc, ac);
    B1.y = tcoef(ln, 5 + 2 * half, bc, ac);
    B2.x = tcoef(ln, 8 + 2 * half, bc, ac);
    B2.y = tcoef(ln, 9 + 2 * half, bc, ac);

    __syncthreads();

#pragma unroll
    for (int t = 0; t < TILES_PER_WAVE; ++t) {
        const long long tbase = wbase + t * TILE;
        const float* tv8 = ldsv + t * 128 + ln * 8;   // staged state of sample `ln`

        // A operand 16x4 layout: lane = M (sample), VGPR j holds K = j + 2*half.
        // K chunks of features: [x,v0,v1,v2], [v3..v6], [v7,0,0,0]
        float xr = (half == 0) ? x_in[tbase + ln] : 0.0f;
        v2f A0, A1, A2;
        A0.x = half ? tv8[1] : xr;
        A0.y = half ? tv8[2] : tv8[0];
        A1.x = tv8[3 + 2 * half];
        A1.y = tv8[4 + 2 * half];
        A2.x = half ? 0.0f : tv8[7];
        A2.y = 0.0f;

        v8f acc = {};
        acc = __builtin_amdgcn_wmma_f32_16x16x4_f32(false, A0, false, B0,
                                                    (short)0, acc, false, false);
        acc = __builtin_amdgcn_wmma_f32_16x16x4_f32(false, A1, false, B1,
                                                    (short)0, acc, false, false);
        acc = __builtin_amdgcn_wmma_f32_16x16x4_f32(false, A2, false, B2,
                                                    (short)0, acc, false, false);

        __syncthreads();   // previous tile's readback finished before overwrite

        // D layout: VGPR r = samples (r, r+8), lane = output feature.
        // Scatter to LDS sample-major: slot 0 = out, slots 4..11 = v_new[0..7],
        // slots 12..18 absorb the zero pad features. Row stride 20 keeps v_new 16B-aligned.
        {
            const int slot = (ln == 0) ? 0 : (ln + 3);
#pragma unroll
            for (int r = 0; r < 8; ++r) {
                ldso[(r + 8 * half) * 20 + slot] = acc[r];
            }
        }
        __syncthreads();

        // Stream results to HBM (non-temporal: read-once/write-once, >> L2 capacity)
        if (l < 16) {
            __builtin_nontemporal_store(ldso[l * 20], &out[tbase + l]);
        }
        {
            const int s = l >> 1, h = l & 1;
            v4f val = *(const v4f*)&ldso[s * 20 + 4 + 4 * h];
            __builtin_nontemporal_store(val, (v4f*)&vnew[(tbase + s) * 8 + 4 * h]);
        }
    }
}

extern "C" void kernel_launch(void* const* d_in, const int* in_sizes, int n_in,
                              void* d_out, int out_size, void* d_ws, size_t ws_size,
                              hipStream_t stream) {
    (void)n_in; (void)out_size; (void)d_ws; (void)ws_size;
    const float* x = (const float*)d_in[0];   // [B,C,1]
    const float* v = (const float*)d_in[1];   // [B,C,8]
    const float* b = (const float*)d_in[2];   // [9]
    const float* a = (const float*)d_in[3];   // [8]
    const int nsamp = in_sizes[0];            // B*C = 8388608
    float* out  = (float*)d_out;              // [B*C]
    float* vnew = out + nsamp;                // [B*C*8]
    const int blocks = nsamp / SAMPLES_PER_BLOCK;  // 16384
    dtdf_wmma_kernel<<<blocks, 256, 0, stream>>>(x, v, b, a, out, vnew);
}